// FkMigrationLinear_11733850652820
// MI455X (gfx1250) — compile-verified
//
#include <hip/hip_runtime.h>
#include <math.h>

// ---- geometry constants ----
#define NDET 128
#define NT   2048
#define NF   1025          // rfft bins
#define NFP  1028          // padded to multiple of 4 (WMMA K-step)
#define NKS  257           // NFP/4 K-steps
#define NX   128
#define NY   256
#define NBC  16            // B*C = 8*2
#define PITCHc 3.0e-4f
#define DTc    2.5e-8f
#define CSND   1540.0f
#define X0c    0.0f
#define DXc    3.0e-4f
#define Y0c    1.0e-3f
#define DYc    1.5e-4f
#define EPSF   1.1920929e-7f
#define PI_F   3.14159265358979323846f

typedef __attribute__((ext_vector_type(2))) float v2f;
typedef __attribute__((ext_vector_type(8))) float v8f;

// Vector pointee types matching the async-LDS builtin signatures exactly
// (clang diagnostic: "__attribute__((__vector_size__(4*sizeof(int)))) int __device__ *")
typedef __attribute__((__vector_size__(4 * sizeof(int)))) int vsi4;
typedef __attribute__((__vector_size__(2 * sizeof(int)))) int vsi2;

// ---- gfx1250 async global->LDS staging (ASYNCcnt path) ----
#if defined(__has_builtin)
#if __has_builtin(__builtin_amdgcn_global_load_async_to_lds_b128) && \
    __has_builtin(__builtin_amdgcn_global_load_async_to_lds_b64)
#define HAVE_ASYNC_LDS 1
#endif
#endif
#ifndef HAVE_ASYNC_LDS
#define HAVE_ASYNC_LDS 0
#endif

// Integer round-trip addrspace casts: on AMDGPU a flat pointer to LDS carries
// the LDS offset in its low 32 bits, and a flat global pointer's value equals
// the global address, so these are value-correct.
#define G128(p) ((__attribute__((address_space(1))) vsi4*)(unsigned long long)(uintptr_t)(p))
#define L128(p) ((__attribute__((address_space(3))) vsi4*)(unsigned int)(uintptr_t)(p))
#define G64(p)  ((__attribute__((address_space(1))) vsi2*)(unsigned long long)(uintptr_t)(p))
#define L64(p)  ((__attribute__((address_space(3))) vsi2*)(unsigned int)(uintptr_t)(p))

static __device__ __forceinline__ void async_copy_b128(const void* g, void* l) {
#if HAVE_ASYNC_LDS
    __builtin_amdgcn_global_load_async_to_lds_b128(G128(g), L128(l), 0, 0);
#else
    *(float4*)l = *(const float4*)g;
#endif
}
static __device__ __forceinline__ void async_copy_b64(const void* g, void* l) {
#if HAVE_ASYNC_LDS
    __builtin_amdgcn_global_load_async_to_lds_b64(G64(g), L64(l), 0, 0);
#else
    *(float2*)l = *(const float2*)g;
#endif
}
static __device__ __forceinline__ void async_wait_all() {
#if HAVE_ASYNC_LDS
#if __has_builtin(__builtin_amdgcn_s_wait_asynccnt)
    __builtin_amdgcn_s_wait_asynccnt(0);
#else
    asm volatile("s_wait_asynccnt 0" ::: "memory");
#endif
#endif
}

static __device__ __forceinline__ float kx_of(int d) {
    // 2*pi*fftfreq(128, pitch)
    float fr = (d < 64) ? (float)d : (float)(d - 128);
    return 2.0f * PI_F * fr / (128.0f * PITCHc);
}

static __device__ __forceinline__ v8f cwmma(v2f a, v2f b, v8f c) {
    // D = A(16x4,f32) * B(4x16,f32) + C  -> v_wmma_f32_16x16x4_f32
    return __builtin_amdgcn_wmma_f32_16x16x4_f32(false, a, false, b, (short)0, c,
                                                 false, false);
}

// spec pair-interleaved layout: element (d, f, bc), f = 4*ks + 2*kh + j
//   flat = ((((d*NKS + ks)*2 + kh)*NBC + bc)*2 + j
static __device__ __forceinline__ size_t spec_idx(int d, int f, int bc) {
    int ks = f >> 2, kh = (f >> 1) & 1, j = f & 1;
    return ((((size_t)d * NKS + ks) * 2 + kh) * NBC + bc) * 2 + j;
}
// band pair-interleaved layout: element (d, y, bc)
//   flat = (((y*64 + d/2)*NBC + bc)*2 + (d&1)
static __device__ __forceinline__ size_t band_idx(int d, int y, int bc) {
    return (((size_t)y * 64 + (d >> 1)) * NBC + bc) * 2 + (d & 1);
}

// ---------------- K0a: apod sum -> inv_norm (stats[32]) ----------------
__global__ void k_norm(const float* __restrict__ apod, float* __restrict__ stats) {
    __shared__ float red[128];
    int t = threadIdx.x;
    red[t] = apod[t];
    __syncthreads();
    for (int s = 64; s > 0; s >>= 1) {
        if (t < s) red[t] += red[t + s];
        __syncthreads();
    }
    if (t == 0) {
        float asum = fmaxf(red[0], EPSF);
        stats[32] = 1.0f / (asum * 2048.0f);   // sum(FREQ_W) = 1 + 2*1023 + 1 = 2048
    }
}

// ---------------- K0b: kz table [NDET][NFP] (0 in evanescent/pad region) ----
__global__ void k_kz(float* __restrict__ kz) {
    int idx = blockIdx.x * blockDim.x + threadIdx.x;
    if (idx >= NDET * NFP) return;
    int d = idx / NFP, f = idx % NFP;
    float v = 0.0f;
    if (f < NF) {
        float om = 2.0f * PI_F * (float)f / ((float)NT * DTc);
        float kx = kx_of(d);
        float s  = (om / CSND) * (om / CSND) - kx * kx;
        v = (s > 0.0f) ? sqrtf(s) : 0.0f;
    }
    kz[idx] = v;
}

// ---------------- K0c: phase_x tables, pair-interleaved [d/2][x][2] --------
__global__ void k_px(float* __restrict__ pxr, float* __restrict__ pxi) {
    int idx = blockIdx.x * blockDim.x + threadIdx.x;
    if (idx >= NDET * NX) return;
    int d = idx / NX, x = idx % NX;
    float ang = kx_of(d) * (X0c + (float)x * DXc);
    float s, c;
    __sincosf(ang, &s, &c);
    size_t o = (((size_t)(d >> 1) * NX) + x) * 2 + (d & 1);
    pxr[o] = c;
    pxi[o] = s;
}

// ---------------- K1: per-(b,c) mean / inv-std ----------------
__global__ void k_stats(const float* __restrict__ sino, float* __restrict__ stats) {
    int bc = blockIdx.x;
    int t  = threadIdx.x;
    const float* p = sino + (size_t)bc * NDET * NT;
    float s = 0.0f, s2 = 0.0f;
    for (int i = t; i < NDET * NT; i += 256) {
        float v = p[i];
        s  += v;
        s2 += v * v;
    }
    __shared__ float r1[256], r2[256];
    r1[t] = s; r2[t] = s2;
    __syncthreads();
    for (int k = 128; k > 0; k >>= 1) {
        if (t < k) { r1[t] += r1[t + k]; r2[t] += r2[t + k]; }
        __syncthreads();
    }
    if (t == 0) {
        const float inv_n = 1.0f / (float)(NDET * NT);
        float mean = r1[0] * inv_n;
        float var  = r2[0] * inv_n - mean * mean;
        stats[bc * 2 + 0] = mean;
        stats[bc * 2 + 1] = rsqrtf(var + EPSF);
    }
}

// ---------------- K3: temporal FFT (len 2048, radix-2 DIT in LDS) ----------
// Row staged global->LDS with async B128 copies; standardize+apodize fused.
// Output layout tmp[bc][f][d] (float2).
__global__ void k_fft_time(const float* __restrict__ sino,
                           const float* __restrict__ apod,
                           const float* __restrict__ stats,
                           float2* __restrict__ tmp) {
    __shared__ float2 buf[NT];                     // 16 KB
    __shared__ __align__(16) float stage[NT];      // 8 KB async landing zone
    int row = blockIdx.x;               // bc*128 + d
    int bc = row >> 7, d = row & 127;
    int t  = threadIdx.x;               // 256 threads
    float mean = stats[bc * 2 + 0], istd = stats[bc * 2 + 1];
    float ap = apod[d];
    const float* src = sino + ((size_t)bc * NDET + d) * NT;

    async_copy_b128(src + 4 * t,        stage + 4 * t);          // floats 0..1023
    async_copy_b128(src + 1024 + 4 * t, stage + 1024 + 4 * t);   // floats 1024..2047
    async_wait_all();
    __syncthreads();

    for (int j = t; j < NT; j += 256) {
        unsigned r = __brev((unsigned)j) >> 21;     // 11-bit reverse
        buf[r] = make_float2((stage[j] - mean) * istd * ap, 0.0f);
    }
    __syncthreads();
    for (int s = 1; s <= 11; ++s) {
        int half = 1 << (s - 1);
        for (int b = t; b < NT / 2; b += 256) {
            int pos = b & (half - 1);
            int i0  = ((b >> (s - 1)) << s) + pos;
            int i1  = i0 + half;
            float ang = -PI_F * (float)pos / (float)half;   // -2*pi*pos/(2*half)
            float sn, cs;
            __sincosf(ang, &sn, &cs);
            float2 u = buf[i0], v = buf[i1];
            float2 w = make_float2(v.x * cs - v.y * sn, v.x * sn + v.y * cs);
            buf[i0] = make_float2(u.x + w.x, u.y + w.y);
            buf[i1] = make_float2(u.x - w.x, u.y - w.y);
        }
        __syncthreads();
    }
    for (int j = t; j < NF; j += 256)
        tmp[((size_t)bc * NFP + j) * NDET + d] = buf[j];
}

// ---------------- K4: detector FFT (len 128) + mask + freq weight ----------
// Input tile staged via async B64; writes spec planes in WMMA-B pair layout.
__global__ void k_fft_det(const float2* __restrict__ tmp,
                          const float* __restrict__ kz,
                          float* __restrict__ sr, float* __restrict__ si) {
    int blk = blockIdx.x;               // bc*NFP + f
    int bc = blk / NFP, f = blk % NFP;
    int t  = threadIdx.x;               // 128 threads; t = detector index
    if (f >= NF) {                      // zero pad region (block-uniform branch)
        size_t o = spec_idx(t, f, bc);
        sr[o] = 0.0f; si[o] = 0.0f;
        return;
    }
    __shared__ __align__(16) float2 stage[NDET];   // 1 KB async landing zone
    __shared__ float2 buf[NDET];                   // 1 KB
    const float2* srcrow = tmp + ((size_t)bc * NFP + f) * NDET;
    async_copy_b64(srcrow + t, stage + t);
    async_wait_all();
    __syncthreads();

    unsigned r = __brev((unsigned)t) >> 25;   // 7-bit reverse
    buf[r] = stage[t];
    __syncthreads();
    for (int s = 1; s <= 7; ++s) {
        int half = 1 << (s - 1);
        if (t < 64) {
            int pos = t & (half - 1);
            int i0  = ((t >> (s - 1)) << s) + pos;
            int i1  = i0 + half;
            float ang = -PI_F * (float)pos / (float)half;
            float sn, cs;
            __sincosf(ang, &sn, &cs);
            float2 u = buf[i0], v = buf[i1];
            float2 w = make_float2(v.x * cs - v.y * sn, v.x * sn + v.y * cs);
            buf[i0] = make_float2(u.x + w.x, u.y + w.y);
            buf[i1] = make_float2(u.x - w.x, u.y - w.y);
        }
        __syncthreads();
    }
    float2 v   = buf[t];
    float kzv  = kz[t * NFP + f];
    float fw   = ((f == 0) || (f == NF - 1)) ? 1.0f : 2.0f;
    float m    = (kzv > 0.0f) ? fw : 0.0f;
    size_t o   = spec_idx(t, f, bc);
    sr[o] = v.x * m;
    si[o] = v.y * m;
}

// ---------------- K5: band[bc,y,d] = sum_f spec * exp(i*y*kz)  (WMMA) ------
// One wave per (d, 64-row y-group). M=16 y rows, N=16 (b,c) pairs, K=freq.
// Complex GEMM = 4 real v_wmma_f32_16x16x4_f32 per K-step per y-tile.
// Pair-interleaved spec layout -> single b64 load per operand plane.
__global__ void __launch_bounds__(32) k_band(const float* __restrict__ sr,
                                             const float* __restrict__ si,
                                             const float* __restrict__ kz,
                                             float* __restrict__ br,
                                             float* __restrict__ bi) {
    int d    = blockIdx.x & 127;
    int yg   = blockIdx.x >> 7;         // 0..3 (64 y-rows each)
    int lane = threadIdx.x;             // wave32
    __shared__ __align__(16) float kzrow[NFP];     // 4 KB, per-d kz row
    const float* kzg = kz + (size_t)d * NFP;
    for (int j = lane * 4; j < 1024; j += 128)     // 8 async b128 issues/wave
        async_copy_b128(kzg + j, kzrow + j);
    if (lane < 4) kzrow[1024 + lane] = kzg[1024 + lane];
    async_wait_all();
    __syncthreads();

    int kh = lane >> 4;                 // K-half select (VGPR striping)
    int n  = lane & 15;                 // N col (bc) for B; M row (y) for A
    float ybase = Y0c + (float)(yg * 64 + n) * DYc;
    const float ystep = 16.0f * DYc;

    v8f accr[4] = {};                   // band real, 4 y-tiles
    v8f acci[4] = {};                   // band imag

    // pair-interleaved spec: float2 per (ks, kh, bc)
    const float* srd = sr + (((size_t)d * NKS) * 2) * NBC * 2;
    const float* sid = si + (((size_t)d * NKS) * 2) * NBC * 2;

    for (int ks = 0; ks < NKS; ++ks) {
        int f0 = ks << 2;
        int fA = f0 + 2 * kh, fB = fA + 1;
        float kzA = kzrow[fA], kzB = kzrow[fB];

        size_t ob = (((size_t)ks * 2 + kh) * NBC + n) * 2;
        float2 brv = *(const float2*)(srd + ob);
        float2 biv = *(const float2*)(sid + ob);
        v2f Br; Br.x = brv.x; Br.y = brv.y;
        v2f Bi; Bi.x = biv.x; Bi.y = biv.y;
        v2f Bin = -Bi;             // f32 WMMA has no A/B negate (NEG=CNeg only)

        if (ks + 4 < NKS) {        // prefetch future K-tiles toward near cache
            size_t op = (((size_t)(ks + 4) * 2 + kh) * NBC + n) * 2;
            __builtin_prefetch(srd + op, 0, 3);
            __builtin_prefetch(sid + op, 0, 3);
        }

        float sA, cA, sB, cB, wsA, wcA, wsB, wcB;
        __sincosf(ybase * kzA, &sA, &cA);
        __sincosf(ybase * kzB, &sB, &cB);
        __sincosf(ystep * kzA, &wsA, &wcA);   // tile-to-tile rotation
        __sincosf(ystep * kzB, &wsB, &wcB);

#pragma unroll
        for (int tt = 0; tt < 4; ++tt) {
            v2f Ar; Ar.x = cA; Ar.y = cB;
            v2f Ai; Ai.x = sA; Ai.y = sB;
            accr[tt] = cwmma(Ar, Br,  accr[tt]);
            accr[tt] = cwmma(Ai, Bin, accr[tt]);
            acci[tt] = cwmma(Ar, Bi,  acci[tt]);
            acci[tt] = cwmma(Ai, Br,  acci[tt]);
            // rotate phase to next y-tile: P *= exp(i*16*dy*kz)
            float tA = cA * wcA - sA * wsA; sA = cA * wsA + sA * wcA; cA = tA;
            float tB = cB * wcB - sB * wsB; sB = cB * wsB + sB * wcB; cB = tB;
        }
    }

    // Store D tiles into pair-interleaved band layout (matches K6 A loads)
#pragma unroll
    for (int tt = 0; tt < 4; ++tt) {
#pragma unroll
        for (int i = 0; i < 8; ++i) {
            int m = i + 8 * kh;                       // row in 16x16 D tile
            int y = yg * 64 + tt * 16 + m;
            size_t o = band_idx(d, y, n);
            br[o] = accr[tt][i];
            bi[o] = acci[tt][i];
        }
    }
}

// ---------------- K6: img = |band @ phase_x| / norm  (WMMA over d) ---------
// One wave per (y, 16-wide x tile). M=16 (b,c), N=16 x cols, K=d (128).
__global__ void __launch_bounds__(32) k_image(const float* __restrict__ br,
                                              const float* __restrict__ bi,
                                              const float* __restrict__ pxr,
                                              const float* __restrict__ pxi,
                                              const float* __restrict__ stats,
                                              float* __restrict__ out) {
    int y    = blockIdx.x & 255;
    int xt   = blockIdx.x >> 8;        // 0..7
    int lane = threadIdx.x;
    int kh = lane >> 4, n = lane & 15;
    int xc = xt * 16 + n;
    float inv_norm = stats[32];

    v8f cr = {}, ci = {};
    for (int d0 = 0; d0 < NDET; d0 += 4) {
        int p = (d0 >> 1) + kh;                      // d-pair index
        size_t oa = (((size_t)y * 64 + p) * NBC + n) * 2;
        float2 arv = *(const float2*)(br + oa);
        float2 aiv = *(const float2*)(bi + oa);
        v2f Ar; Ar.x = arv.x; Ar.y = arv.y;
        v2f Ai; Ai.x = aiv.x; Ai.y = aiv.y;
        size_t obp = ((size_t)p * NX + xc) * 2;
        float2 brv = *(const float2*)(pxr + obp);
        float2 biv = *(const float2*)(pxi + obp);
        v2f Br; Br.x = brv.x; Br.y = brv.y;
        v2f Bi; Bi.x = biv.x; Bi.y = biv.y;
        v2f Bin = -Bi;
        cr = cwmma(Ar, Br,  cr);
        cr = cwmma(Ai, Bin, cr);
        ci = cwmma(Ar, Bi,  ci);
        ci = cwmma(Ai, Br,  ci);
    }
#pragma unroll
    for (int i = 0; i < 8; ++i) {
        int m = i + 8 * kh;                            // (b,c) index
        float re = cr[i], im = ci[i];
        out[((size_t)m * NY + y) * NX + xt * 16 + n] =
            sqrtf(re * re + im * im) * inv_norm;
    }
}

// ---------------- launch ----------------
extern "C" void kernel_launch(void* const* d_in, const int* in_sizes, int n_in,
                              void* d_out, int out_size, void* d_ws, size_t ws_size,
                              hipStream_t stream) {
    (void)in_sizes; (void)n_in; (void)out_size; (void)ws_size;
    const float* sino = (const float*)d_in[0];   // [8,2,128,2048]
    const float* apod = (const float*)d_in[1];   // [128]
    float* out = (float*)d_out;                  // [8,2,256,128]
    float* ws  = (float*)d_ws;

    size_t off = 0;
    float* stats = ws + off;  off += 64;                          // mean/istd pairs + inv_norm
    float* kz    = ws + off;  off += (size_t)NDET * NFP;          // 0.53 MB
    float* pxr   = ws + off;  off += (size_t)NDET * NX;
    float* pxi   = ws + off;  off += (size_t)NDET * NX;
    float2* tmp  = (float2*)(ws + off); off += (size_t)NBC * NFP * NDET * 2;  // 16.8 MB
    float* sr    = ws + off;  off += (size_t)NDET * NKS * 2 * NBC * 2;  // 8.4 MB
    float* si    = ws + off;  off += (size_t)NDET * NKS * 2 * NBC * 2;  // 8.4 MB
    float* br    = ws + off;  off += (size_t)NY * 64 * NBC * 2;   // 2.1 MB
    float* bi    = ws + off;  off += (size_t)NY * 64 * NBC * 2;   // 2.1 MB

    k_norm    <<<1, 128, 0, stream>>>(apod, stats);
    k_kz      <<<(NDET * NFP + 255) / 256, 256, 0, stream>>>(kz);
    k_px      <<<(NDET * NX + 255) / 256, 256, 0, stream>>>(pxr, pxi);
    k_stats   <<<NBC, 256, 0, stream>>>(sino, stats);
    k_fft_time<<<NBC * NDET, 256, 0, stream>>>(sino, apod, stats, tmp);
    k_fft_det <<<NBC * NFP, 128, 0, stream>>>(tmp, kz, sr, si);
    k_band    <<<NDET * 4, 32, 0, stream>>>(sr, si, kz, br, bi);
    k_image   <<<NY * 8, 32, 0, stream>>>(br, bi, pxr, pxi, stats, out);
}